// GraphSAGELayer_43679817400489
// MI455X (gfx1250) — compile-verified
//
#include <hip/hip_runtime.h>

#define N_NODES 50000
#define N_EDGES 640000
#define IN_CH   128
#define OUT_CH  128
#define CAT_CH  256          // 2*IN_CH
#define LDS_STRIDE 260       // 256 + 4-float skew -> conflict-free column access

typedef float v2f __attribute__((ext_vector_type(2)));
typedef float v8f __attribute__((ext_vector_type(8)));

// ---------------- Kernel 1: zero agg + deg in workspace ----------------
__global__ void zero_ws_kernel(float* __restrict__ ws, int n) {
    int i = blockIdx.x * blockDim.x + threadIdx.x;
    if (i < n) ws[i] = 0.0f;
}

// ---------------- Kernel 2: scatter-add neighbor features --------------
// 32 lanes per edge; each lane handles 4 consecutive channels (float4).
// unsafeAtomicAdd guarantees lowering to GLOBAL_ATOMIC_ADD_F32 (no CAS loop).
__global__ void scatter_edges_kernel(const float* __restrict__ x,
                                     const long long* __restrict__ ei,  // [2, N_EDGES] int64
                                     float* __restrict__ agg,
                                     float* __restrict__ deg) {
    int tid = blockIdx.x * blockDim.x + threadIdx.x;
    int e = tid >> 5;          // edge id
    int l = tid & 31;          // channel group (4 ch each)
    if (e >= N_EDGES) return;

    long long row = ei[e];              // destination (segment id)
    long long col = ei[N_EDGES + e];    // source (gathered)

    const float4 v = *(const float4*)(x + col * IN_CH + l * 4);
    float* dst = agg + row * IN_CH + l * 4;
    unsafeAtomicAdd(dst + 0, v.x);
    unsafeAtomicAdd(dst + 1, v.y);
    unsafeAtomicAdd(dst + 2, v.z);
    unsafeAtomicAdd(dst + 3, v.w);
    if (l == 0) unsafeAtomicAdd(deg + row, 1.0f);
}

// ---------------- Kernel 3: normalize + concat + GEMM (WMMA f32) -------
// One wave32 per 16-node tile. out[16,128] = cat([x, agg/deg])[16,256] @ W^T + b
// k-outer / n-inner: 8 independent accumulator chains -> 8 hazard-free WMMAs
// per k-step, A fragment read from LDS once per k-step.
__global__ __launch_bounds__(32) void sage_gemm_kernel(
        const float* __restrict__ x,
        const float* __restrict__ agg,
        const float* __restrict__ deg,
        const float* __restrict__ W,     // [OUT_CH, CAT_CH] row-major
        const float* __restrict__ bias,  // [OUT_CH]
        float* __restrict__ out) {       // [N_NODES, OUT_CH]
    __shared__ float lds[16 * LDS_STRIDE];
    __shared__ float rdeg[16];

    const int lane   = threadIdx.x;      // 0..31
    const int m_base = blockIdx.x * 16;  // node tile base (50000 = 16*3125 exact)

    if (lane < 16) {
        float d = deg[m_base + lane];
        rdeg[lane] = 1.0f / fmaxf(d, 1.0f);
    }
    __syncthreads();

    // Stage concat([x, agg*rdeg]) tile [16, CAT_CH] into LDS (uniform trip count).
    for (int i = lane; i < 16 * CAT_CH; i += 32) {
        int r = i >> 8;          // row 0..15
        int c = i & 255;         // col 0..255
        int node = m_base + r;
        float v;
        if (c < IN_CH) v = x[(long long)node * IN_CH + c];
        else           v = agg[(long long)node * IN_CH + (c - IN_CH)] * rdeg[r];
        lds[r * LDS_STRIDE + c] = v;
    }
    __syncthreads();

    // WMMA f32 16x16x4 fragment indexing (ISA 7.12.2):
    //   A: lane L -> row M = L%16, VGPR0/1 hold K = 2*(L/16), 2*(L/16)+1
    //   B: lane L -> col N = L%16, VGPR0/1 hold K = 2*(L/16), 2*(L/16)+1
    //   D: VGPR r -> row M = r + 8*(L/16), col N = L%16
    const int nlo   = lane & 15;
    const int khalf = (lane >> 4) * 2;   // 0 or 2
    const int mhi   = (lane >> 4) * 8;   // 0 or 8

    v8f acc[8];
    #pragma unroll
    for (int nt = 0; nt < 8; ++nt) {
        const float bn = bias[nt * 16 + nlo];
        acc[nt] = (v8f){bn, bn, bn, bn, bn, bn, bn, bn};
    }

    // B[k][n] = W[n][k]; lane's row of tile nt lives at W + (nt*16 + nlo)*CAT_CH
    const float* wbase = W + (long long)nlo * CAT_CH + khalf;
    const float* arow  = &lds[nlo * LDS_STRIDE + khalf];

    for (int k0 = 0; k0 < CAT_CH; k0 += 4) {
        const v2f a = *(const v2f*)(arow + k0);          // 8B-aligned ds load
        #pragma unroll
        for (int nt = 0; nt < 8; ++nt) {
            const v2f b = *(const v2f*)(wbase + nt * 16 * CAT_CH + k0);
            acc[nt] = __builtin_amdgcn_wmma_f32_16x16x4_f32(
                    /*neg_a=*/false, a, /*neg_b=*/false, b,
                    /*c_mod=*/(short)0, acc[nt],
                    /*reuse_a=*/false, /*reuse_b=*/false);
        }
    }

    #pragma unroll
    for (int nt = 0; nt < 8; ++nt) {
        #pragma unroll
        for (int r = 0; r < 8; ++r) {
            out[(long long)(m_base + mhi + r) * OUT_CH + nt * 16 + nlo] = acc[nt][r];
        }
    }
}

extern "C" void kernel_launch(void* const* d_in, const int* in_sizes, int n_in,
                              void* d_out, int out_size, void* d_ws, size_t ws_size,
                              hipStream_t stream) {
    const float*     x  = (const float*)d_in[0];       // [N_NODES, IN_CH] f32
    const long long* ei = (const long long*)d_in[1];   // [2, N_EDGES] int64
    const float*     W  = (const float*)d_in[2];       // [OUT_CH, 2*IN_CH] f32
    const float*     b  = (const float*)d_in[3];       // [OUT_CH] f32
    float*           out = (float*)d_out;              // [N_NODES, OUT_CH] f32

    float* agg = (float*)d_ws;                             // N_NODES*IN_CH floats
    float* deg = agg + (size_t)N_NODES * IN_CH;            // N_NODES floats

    // 1) zero workspace (must happen every call; harness does not re-poison)
    int nzero = N_NODES * IN_CH + N_NODES;
    zero_ws_kernel<<<(nzero + 255) / 256, 256, 0, stream>>>((float*)d_ws, nzero);

    // 2) scatter-add x[col] into agg[row], count degrees
    int nscatter = N_EDGES * 32;
    scatter_edges_kernel<<<(nscatter + 255) / 256, 256, 0, stream>>>(x, ei, agg, deg);

    // 3) fused normalize + concat + GEMM with f32 WMMA
    sage_gemm_kernel<<<N_NODES / 16, 32, 0, stream>>>(x, agg, deg, W, b, out);
}